// Transformer_4432406250083
// MI455X (gfx1250) — compile-verified
//
#include <hip/hip_runtime.h>

// Tiny fused transformer for MI455X (gfx1250, wave32).
// DIM=4, VOC=10, SEQ=8, NUM_LAYER=3, BATCH=524288.
// Reference MLP has NO nonlinearity -> everything between softmaxes is affine,
// pre-composed once per block:
//   Tt[10][16]   token -> layer-1 qkv table
//   bridge l: M (4x16) + c (16): attention_l output -> next qkv / logits
// Softmax: no max-subtract, no shuffles, no divides. q-columns carry
// 0.5*log2(e) so logits are already in exp2 domain -> raw v_exp_f32; the
// denominator is produced inside the attn@v WMMA via a ones-column
// ([V | 1]); the next projection scales its A-operand by rcp(rowsum).
// Padding columns are structurally zero -> no conditional LDS loads.
// Each wave processes NITER=4 pairs of batch items to amortize staging.

typedef float v2f __attribute__((ext_vector_type(2)));
typedef float v8f __attribute__((ext_vector_type(8)));

#define LP 17        // staging tile pitch (16+1)
#define NITER 4      // item-pairs per wave
#define QSCALE 0.72134752044448170368f  // 0.5 * log2(e): SCALE + exp2 folding

__device__ __forceinline__ v8f wmma_k4(v2f a, v2f b, v8f c) {
  return __builtin_amdgcn_wmma_f32_16x16x4_f32(
      false, a, false, b, (short)0, c, false, false);
}

__device__ __forceinline__ v8f splat8(float v) {
  v8f r;
#pragma unroll
  for (int j = 0; j < 8; ++j) r[j] = v;
  return r;
}

// Store a 16x16 f32 D-matrix into the per-wave LDS tile S[m][n].
__device__ __forceinline__ void stage_D(float* S, int lane, v8f d) {
  const int h = lane >> 4;
  const int n = lane & 15;
#pragma unroll
  for (int j = 0; j < 8; ++j) S[(j + 8 * h) * LP + n] = d[j];
  __asm__ volatile("" ::: "memory");  // LDS is wave-in-order; stop reordering
}

// Block-diag attention over a qkv source (rows = 16 tokens, cols 0..3 = q
// (pre-scaled by QSCALE), 4..7 = k, 8..11 = v, 12..15 = 0). Returns
// UN-normalized av_aug: numerator in cols tb..tb+3, row-sum in col tb+4.
__device__ __forceinline__ v8f attention_aug(
    float* S, int lane, const float* rq, const float* rv0, const float* rv1,
    const float* rv2, const float* rv3) {
  const int h = lane >> 4, n = lane & 15, kp = h * 2, tb = (n >> 3) * 8;
  v2f qa, kb;
  qa.x = rq[kp];
  qa.y = rq[kp + 1];
  kb.x = rq[4 + kp];
  kb.y = rq[4 + kp + 1];
  const int vc = 8 + (n & 7);                     // vc >= 12 reads zero padding
  const float ones = ((n & 7) == 4) ? 1.f : 0.f;  // denominator column
  v2f vb1, vb2;                                   // [V | 1 | 0] in B-layout
  vb1.x = rv0[vc] + ones;
  vb1.y = rv1[vc] + ones;
  vb2.x = rv2[vc] + ones;
  vb2.y = rv3[vc] + ones;

  v8f logit = wmma_k4(qa, kb, splat8(0.f));  // already in exp2 domain
  v8f e;
#pragma unroll
  for (int j = 0; j < 8; ++j) e[j] = __builtin_amdgcn_exp2f(logit[j]);
  stage_D(S, lane, e);
  v2f a1, a2;  // e in A-layout, tokens kp..kp+1 / kp+4..kp+5 of own block
  a1.x = S[n * LP + tb + kp];
  a1.y = S[n * LP + tb + kp + 1];
  a2.x = S[n * LP + tb + 4 + kp];
  a2.y = S[n * LP + tb + 4 + kp + 1];
  v8f av = wmma_k4(a1, vb1, splat8(0.f));
  return wmma_k4(a2, vb2, av);
}

// Fused affine bridge: normalizes staged av_aug by rcp(rowsum) on the
// A-operand, then applies Mc = [M 4x16 | c 16].
__device__ __forceinline__ v8f proj_norm(float* S, int lane, const float* Mc) {
  const int h = lane >> 4, n = lane & 15, kp = h * 2, tb = (n >> 3) * 8;
  const float r = __builtin_amdgcn_rcpf(S[n * LP + tb + 4]);
  v2f a, b;
  a.x = S[n * LP + tb + kp] * r;
  a.y = S[n * LP + tb + kp + 1] * r;
  b.x = Mc[kp * 16 + n];
  b.y = Mc[(kp + 1) * 16 + n];
  return wmma_k4(a, b, splat8(Mc[64 + n]));
}

__global__ __launch_bounds__(256) void tiny_transformer_kernel(
    const int* __restrict__ x, const float* __restrict__ emb,
    const float* __restrict__ kqv_w, const float* __restrict__ kqv_b,
    const float* __restrict__ out_w, const float* __restrict__ out_b,
    const float* __restrict__ mlp1_w, const float* __restrict__ mlp1_b,
    const float* __restrict__ mlp2_w, const float* __restrict__ mlp2_b,
    const float* __restrict__ lm_w, const float* __restrict__ lm_b,
    float* __restrict__ out, int nBatch) {
  __shared__ float Tt[10 * 16];       // token -> layer-1 qkv (cols 12..15 = 0)
  __shared__ float fused[3 * 80];     // per bridge: M 4x16 at +0, c at +64
  __shared__ float tmp1[3][4][8];     // Wo@W1
  __shared__ float tb1[3][8];         // bo@W1 + b1
  __shared__ float tmp2[3][4][4];     // (Wo@W1)@W2
  __shared__ float tb2[3][4];         // ((bo@W1+b1)@W2 + b2)
  __shared__ float Sall[8][16 * LP];  // per-wave staging tiles
  __shared__ int tokLds[8][16];

  const int tid = threadIdx.x;

  // ---- pre-compose the affine chain (once per block, 64 items amortize it)
  for (int i = tid; i < 160; i += 256) Tt[i] = 0.f;
  for (int i = tid; i < 240; i += 256) fused[i] = 0.f;
  __syncthreads();
  for (int i = tid; i < 120; i += 256) {  // Tt[v][c] = emb[v]@Wq1 + bq1
    int v = i / 12, c = i % 12;
    float s = kqv_b[c];
    for (int k = 0; k < 4; ++k) s += emb[v * 4 + k] * kqv_w[k * 12 + c];
    Tt[v * 16 + c] = (c < 4 ? QSCALE : 1.f) * s;
  }
  for (int i = tid; i < 96; i += 256) {  // tmp1 = Wo@W1
    int l = i / 32, r = (i % 32) / 8, c = i % 8;
    float s = 0.f;
    for (int k = 0; k < 4; ++k)
      s += out_w[l * 16 + r * 4 + k] * mlp1_w[l * 32 + k * 8 + c];
    tmp1[l][r][c] = s;
  }
  for (int i = tid; i < 24; i += 256) {  // tb1 = bo@W1 + b1
    int l = i / 8, c = i % 8;
    float s = mlp1_b[l * 8 + c];
    for (int k = 0; k < 4; ++k) s += out_b[l * 4 + k] * mlp1_w[l * 32 + k * 8 + c];
    tb1[l][c] = s;
  }
  __syncthreads();
  for (int i = tid; i < 48; i += 256) {  // tmp2 = tmp1@W2
    int l = i / 16, r = (i % 16) / 4, c = i % 4;
    float s = 0.f;
    for (int k = 0; k < 8; ++k) s += tmp1[l][r][k] * mlp2_w[l * 32 + k * 4 + c];
    tmp2[l][r][c] = s;
  }
  for (int i = tid; i < 12; i += 256) {  // tb2 = tb1@W2 + b2
    int l = i / 4, c = i % 4;
    float s = mlp2_b[l * 4 + c];
    for (int k = 0; k < 8; ++k) s += tb1[l][k] * mlp2_w[l * 32 + k * 4 + c];
    tb2[l][c] = s;
  }
  __syncthreads();
  for (int i = tid; i < 96; i += 256) {  // bridges 0,1: M = tmp2@Wq_{l+1}
    int l = i / 48, r = (i % 48) / 12, c = i % 12;
    float s = 0.f;
    for (int k = 0; k < 4; ++k)
      s += tmp2[l][r][k] * kqv_w[(l + 1) * 48 + k * 12 + c];
    fused[l * 80 + r * 16 + c] = (c < 4 ? QSCALE : 1.f) * s;
  }
  for (int i = tid; i < 24; i += 256) {  // bridges 0,1: c = tb2@Wq + bq
    int l = i / 12, c = i % 12;
    float s = kqv_b[(l + 1) * 12 + c];
    for (int k = 0; k < 4; ++k)
      s += tb2[l][k] * kqv_w[(l + 1) * 48 + k * 12 + c];
    fused[l * 80 + 64 + c] = (c < 4 ? QSCALE : 1.f) * s;
  }
  for (int i = tid; i < 40; i += 256) {  // bridge 2: M = tmp2@WL
    int r = i / 10, c = i % 10;
    float s = 0.f;
    for (int k = 0; k < 4; ++k) s += tmp2[2][r][k] * lm_w[k * 10 + c];
    fused[160 + r * 16 + c] = s;
  }
  for (int i = tid; i < 10; i += 256) {  // bridge 2: c = tb2@WL + lm_b
    float s = lm_b[i];
    for (int k = 0; k < 4; ++k) s += tb2[2][k] * lm_w[k * 10 + i];
    fused[160 + 64 + i] = s;
  }
  __syncthreads();

  // ---- per-wave pipeline: 3 attentions + 3 fused projections per item-pair
  const int lane = tid & 31;
  const int wid = tid >> 5;
  float* S = Sall[wid];
  const int n = lane & 15;
  const int kp = (lane >> 4) * 2;
  const int vr = (n >> 3) * 8;

  const float* sq = S + n * LP;
  const float* sv0 = S + (vr + kp) * LP;
  const float* sv1 = S + (vr + kp + 1) * LP;
  const float* sv2 = S + (vr + kp + 4) * LP;
  const float* sv3 = S + (vr + kp + 5) * LP;

  const long long gw = (long long)blockIdx.x * 8 + wid;  // global wave id

#pragma unroll 1
  for (int it = 0; it < NITER; ++it) {
    long long item0 = (gw * NITER + it) * 2;              // 2 items per pass
    if (item0 > (long long)nBatch - 2) item0 = nBatch - 2;  // clamp, no exit

    if (lane < 16)  // row n's token id (rows: item*8 + token)
      tokLds[wid][n] =
          __builtin_nontemporal_load(&x[(item0 + (n >> 3)) * 8 + (n & 7)]);
    __asm__ volatile("" ::: "memory");

    // layer-1 attention straight from the token->qkv table
    const float* rq = Tt + tokLds[wid][n] * 16;
    const float* rv0 = Tt + tokLds[wid][vr + kp] * 16;
    const float* rv1 = Tt + tokLds[wid][vr + kp + 1] * 16;
    const float* rv2 = Tt + tokLds[wid][vr + kp + 4] * 16;
    const float* rv3 = Tt + tokLds[wid][vr + kp + 5] * 16;
    v8f av = attention_aug(S, lane, rq, rv0, rv1, rv2, rv3);
    stage_D(S, lane, av);

    v8f qkv = proj_norm(S, lane, fused + 0);  // -> layer-2 qkv
    stage_D(S, lane, qkv);
    av = attention_aug(S, lane, sq, sv0, sv1, sv2, sv3);
    stage_D(S, lane, av);

    qkv = proj_norm(S, lane, fused + 80);  // -> layer-3 qkv
    stage_D(S, lane, qkv);
    av = attention_aug(S, lane, sq, sv0, sv1, sv2, sv3);
    stage_D(S, lane, av);

    v8f lg = proj_norm(S, lane, fused + 160);  // -> output logits (cols 0..9)
    stage_D(S, lane, lg);

    // coalesced epilogue: 5 contiguous non-temporal stores per wave
    const long long obase = item0 * 80;  // 2 items * 8 tokens * 10 vocab
#pragma unroll
    for (int i = 0; i < 5; ++i) {
      const int f = lane + 32 * i;  // 0..159
      const int m = f / 10;         // item*8 + token
      const int v = f - m * 10;     // vocab index
      __builtin_nontemporal_store(S[m * LP + v], &out[obase + f]);
    }
  }
}

extern "C" void kernel_launch(void* const* d_in, const int* in_sizes, int n_in,
                              void* d_out, int out_size, void* d_ws, size_t ws_size,
                              hipStream_t stream) {
  const int* x = (const int*)d_in[0];
  const float* emb = (const float*)d_in[1];
  const float* kqv_w = (const float*)d_in[2];
  const float* kqv_b = (const float*)d_in[3];
  const float* out_w = (const float*)d_in[4];
  const float* out_b = (const float*)d_in[5];
  const float* mlp1_w = (const float*)d_in[6];
  const float* mlp1_b = (const float*)d_in[7];
  const float* mlp2_w = (const float*)d_in[8];
  const float* mlp2_b = (const float*)d_in[9];
  const float* lm_w = (const float*)d_in[10];
  const float* lm_b = (const float*)d_in[11];
  float* out = (float*)d_out;

  const int nBatch = in_sizes[0] / 8;  // x is [B, SEQ]
  const int itemsPerBlock = 16 * NITER;  // 8 waves * 2 items * NITER
  const int blocks = (nBatch + itemsPerBlock - 1) / itemsPerBlock;

  tiny_transformer_kernel<<<blocks, 256, 0, stream>>>(
      x, emb, kqv_w, kqv_b, out_w, out_b, mlp1_w, mlp1_b, mlp2_w, mlp2_b,
      lm_w, lm_b, out, nBatch);
}